// Selector_50096498540586
// MI455X (gfx1250) — compile-verified
//
#include <hip/hip_runtime.h>
#include <hip/hip_bf16.h>
#include <stdint.h>

// ---------------------------------------------------------------------------
// Selector: q/k projections (WMMA f32 16x16x4) + LN + gumbel-softmax masks +
// bandwidth-bound soft gather over V (float4 streaming, non-temporal).
// ---------------------------------------------------------------------------

typedef float v2f __attribute__((ext_vector_type(2)));
typedef float v4f __attribute__((ext_vector_type(4)));
typedef float v8f __attribute__((ext_vector_type(8)));

#define DIM 512
#define NCLIP 32
#define BS 32
#define TOPK 8
#define DFLAT 131072  // frames*obj_num*obj_dim = 16*16*512

// ------------------------- WMMA fp32 GEMM ----------------------------------
// C[m,n] = sum_j A[m,j] * W[n,j] + bias[n];  A: [M,512], W: [512,512]
// One wave (32 threads) computes a 16x16 tile with V_WMMA_F32_16X16X4_F32.
__global__ void __launch_bounds__(32)
gemm_wmma_f32(const float* __restrict__ A, const float* __restrict__ W,
              const float* __restrict__ bias, float* __restrict__ C) {
  const int lane = threadIdx.x;        // 0..31
  const int half = lane >> 4;          // 0 or 1 (selects K-pair 0,1 vs 2,3)
  const int l    = lane & 15;
  const int m0 = blockIdx.x * 16;
  const int n0 = blockIdx.y * 16;

  // A 16x4 layout: lanes 0-15 M=l hold K=0,1 in v0,v1; lanes 16-31 hold K=2,3
  const float* arow = A + (size_t)(m0 + l) * DIM + 2 * half;
  // B 4x16 layout (W^T tile): per lane fixed N=l, VGPRs hold K pair
  const float* wrow = W + (size_t)(n0 + l) * DIM + 2 * half;

  v8f acc = {0.f, 0.f, 0.f, 0.f, 0.f, 0.f, 0.f, 0.f};
  #pragma unroll 4
  for (int k = 0; k < DIM; k += 4) {
    v2f a = *(const v2f*)(arow + k);
    v2f b = *(const v2f*)(wrow + k);
    acc = __builtin_amdgcn_wmma_f32_16x16x4_f32(
        /*neg_a=*/false, a, /*neg_b=*/false, b,
        /*c_mod=*/(short)0, acc, /*reuse_a=*/false, /*reuse_b=*/false);
  }

  // C/D layout: VGPR v, lanes 0-15 -> row m0+v, lanes 16-31 -> row m0+v+8
  const float bn = bias[n0 + l];
  #pragma unroll
  for (int v = 0; v < 8; ++v)
    C[(size_t)(m0 + v + 8 * half) * DIM + n0 + l] = acc[v] + bn;
}

// ------------------------- LayerNorm (block per row) ------------------------
__global__ void __launch_bounds__(256)
ln_kernel(float* __restrict__ x, const float* __restrict__ gamma,
          const float* __restrict__ beta) {
  const int tid = threadIdx.x;
  float* xr = x + (size_t)blockIdx.x * DIM;
  float v0 = xr[tid], v1 = xr[tid + 256];
  float s = v0 + v1;
  float q = v0 * v0 + v1 * v1;
  #pragma unroll
  for (int o = 16; o; o >>= 1) {
    s += __shfl_xor(s, o, 32);
    q += __shfl_xor(q, o, 32);
  }
  __shared__ float ss[8], sq[8];
  __shared__ float mean_sh, inv_sh;
  const int wave = tid >> 5, lane = tid & 31;
  if (lane == 0) { ss[wave] = s; sq[wave] = q; }
  __syncthreads();
  if (tid == 0) {
    float ts = 0.f, tq = 0.f;
    #pragma unroll
    for (int i = 0; i < 8; ++i) { ts += ss[i]; tq += sq[i]; }
    float mean = ts * (1.0f / DIM);
    float var  = tq * (1.0f / DIM) - mean * mean;
    mean_sh = mean;
    inv_sh  = rsqrtf(var + 1e-12f);
  }
  __syncthreads();
  const float mean = mean_sh, inv = inv_sh;
  xr[tid]       = (v0 - mean) * inv * gamma[tid]       + beta[tid];
  xr[tid + 256] = (v1 - mean) * inv * gamma[tid + 256] + beta[tid + 256];
}

// ------------------------- threefry2x32 (JAX-exact) -------------------------
__device__ __forceinline__ uint32_t rotl32(uint32_t v, uint32_t d) {
  return (v << d) | (v >> (32 - d));
}

__device__ __forceinline__ void threefry2x32(uint32_t k0, uint32_t k1,
                                             uint32_t x0, uint32_t x1,
                                             uint32_t& o0, uint32_t& o1) {
  const uint32_t ks0 = k0, ks1 = k1, ks2 = k0 ^ k1 ^ 0x1BD11BDAu;
  x0 += ks0; x1 += ks1;
  // group 0 (rot A)
  x0 += x1; x1 = rotl32(x1, 13); x1 ^= x0;
  x0 += x1; x1 = rotl32(x1, 15); x1 ^= x0;
  x0 += x1; x1 = rotl32(x1, 26); x1 ^= x0;
  x0 += x1; x1 = rotl32(x1,  6); x1 ^= x0;
  x0 += ks1; x1 += ks2 + 1u;
  // group 1 (rot B)
  x0 += x1; x1 = rotl32(x1, 17); x1 ^= x0;
  x0 += x1; x1 = rotl32(x1, 29); x1 ^= x0;
  x0 += x1; x1 = rotl32(x1, 16); x1 ^= x0;
  x0 += x1; x1 = rotl32(x1, 24); x1 ^= x0;
  x0 += ks2; x1 += ks0 + 2u;
  // group 2 (rot A)
  x0 += x1; x1 = rotl32(x1, 13); x1 ^= x0;
  x0 += x1; x1 = rotl32(x1, 15); x1 ^= x0;
  x0 += x1; x1 = rotl32(x1, 26); x1 ^= x0;
  x0 += x1; x1 = rotl32(x1,  6); x1 ^= x0;
  x0 += ks0; x1 += ks1 + 3u;
  // group 3 (rot B)
  x0 += x1; x1 = rotl32(x1, 17); x1 ^= x0;
  x0 += x1; x1 = rotl32(x1, 29); x1 ^= x0;
  x0 += x1; x1 = rotl32(x1, 16); x1 ^= x0;
  x0 += x1; x1 = rotl32(x1, 24); x1 ^= x0;
  x0 += ks1; x1 += ks2 + 4u;
  // group 4 (rot A)
  x0 += x1; x1 = rotl32(x1, 13); x1 ^= x0;
  x0 += x1; x1 = rotl32(x1, 15); x1 ^= x0;
  x0 += x1; x1 = rotl32(x1, 26); x1 ^= x0;
  x0 += x1; x1 = rotl32(x1,  6); x1 ^= x0;
  x0 += ks2; x1 += ks0 + 5u;
  o0 = x0; o1 = x1;
}

// ------------------- logits -> softmax -> gumbel-softmax masks --------------
// One wave per batch; lane == clip index n.
__global__ void __launch_bounds__(32)
mask_kernel(const float* __restrict__ q, const float* __restrict__ k,
            float* __restrict__ masks) {
  const int b = blockIdx.x;
  const int n = threadIdx.x;  // 0..31
  const float* kr = k + ((size_t)b * NCLIP + n) * DIM;
  const float* qr = q + (size_t)b * DIM;
  float acc = 0.f;
  #pragma unroll 8
  for (int j = 0; j < DIM; ++j) acc += kr[j] * qr[j];

  // softmax over lanes (n_clip == wave32)
  float m = acc;
  #pragma unroll
  for (int o = 16; o; o >>= 1) m = fmaxf(m, __shfl_xor(m, o, 32));
  float e = expf(acc - m);
  float s = e;
  #pragma unroll
  for (int o = 16; o; o >>= 1) s += __shfl_xor(s, o, 32);
  const float prob = e / s;

  const float TINY = 1.1754943508222875e-38f;  // float32 tiny (matches JAX)
  #pragma unroll
  for (int t = 0; t < TOPK; ++t) {
    const uint32_t i = (uint32_t)(t * 1024 + b * 32 + n);  // flat idx in 8192
    uint32_t o0, o1, bits;
    if (i < 4096u) { threefry2x32(0u, 123u, i, i + 4096u, o0, o1); bits = o0; }
    else           { threefry2x32(0u, 123u, i - 4096u, i, o0, o1); bits = o1; }
    float u = __uint_as_float((bits >> 9) | 0x3f800000u) - 1.0f;  // [0,1)
    u = fmaxf(u * (1.0f - TINY) + TINY, TINY);
    const float g = -logf(-logf(u));

    float lt = prob + g;
    float mt = lt;
    #pragma unroll
    for (int o = 16; o; o >>= 1) mt = fmaxf(mt, __shfl_xor(mt, o, 32));
    float et = expf(lt - mt);
    float st = et;
    #pragma unroll
    for (int o = 16; o; o >>= 1) st += __shfl_xor(st, o, 32);
    masks[(size_t)b * (TOPK * NCLIP) + t * NCLIP + n] = et / st;
  }
}

// --------------- soft gather: out[b,t,D] = sum_n masks[b,t,n]*Vf[b,n,D] -----
// Bandwidth-bound: each thread streams one float4 column through all 32 clips,
// keeping 8 float4 accumulators. Non-temporal (V >> 192MB L2, zero reuse).
// Mask weights are kept TRANSPOSED in LDS (sm[n*8+t]) so each n-iteration
// needs only two ds_load_b128 broadcasts instead of eight ds_load_b32.
__global__ void __launch_bounds__(256)
select_kernel(const float* __restrict__ masks, const float* __restrict__ V,
              float* __restrict__ out) {
  __shared__ float sm[NCLIP * TOPK];  // transposed: sm[n*8 + t]
  const int b = blockIdx.y;
  {
    const int i = threadIdx.x;        // i = t*32 + n in source layout
    const int t = i >> 5, n = i & 31;
    sm[n * TOPK + t] = masks[(size_t)b * (TOPK * NCLIP) + i];
  }
  __syncthreads();

  const size_t f = (size_t)blockIdx.x * 256 + threadIdx.x;  // float4 index
  const float* vb = V + (size_t)b * NCLIP * DFLAT + 4 * f;

  v4f acc[TOPK];
  #pragma unroll
  for (int t = 0; t < TOPK; ++t) acc[t] = (v4f){0.f, 0.f, 0.f, 0.f};

  #pragma unroll 4
  for (int n = 0; n < NCLIP; ++n) {
    v4f x = __builtin_nontemporal_load((const v4f*)(vb + (size_t)n * DFLAT));
    const v4f* wp = (const v4f*)(sm + n * TOPK);
    v4f w0 = wp[0];   // weights t=0..3 (one ds_load_b128 broadcast)
    v4f w1 = wp[1];   // weights t=4..7
    acc[0] += w0[0] * x;
    acc[1] += w0[1] * x;
    acc[2] += w0[2] * x;
    acc[3] += w0[3] * x;
    acc[4] += w1[0] * x;
    acc[5] += w1[1] * x;
    acc[6] += w1[2] * x;
    acc[7] += w1[3] * x;
  }

  float* ob = out + (size_t)b * TOPK * DFLAT + 4 * f;
  #pragma unroll
  for (int t = 0; t < TOPK; ++t)
    __builtin_nontemporal_store(acc[t], (v4f*)(ob + (size_t)t * DFLAT));
}

// ---------------------------------------------------------------------------
extern "C" void kernel_launch(void* const* d_in, const int* in_sizes, int n_in,
                              void* d_out, int out_size, void* d_ws, size_t ws_size,
                              hipStream_t stream) {
  const float* Q     = (const float*)d_in[0];   // [32, 512, 1]
  const float* K     = (const float*)d_in[1];   // [32, 32, 512]
  const float* V     = (const float*)d_in[2];   // [32, 32, 16, 16, 512]
  const float* Wq    = (const float*)d_in[3];
  const float* bq    = (const float*)d_in[4];
  const float* gq    = (const float*)d_in[5];
  const float* betaq = (const float*)d_in[6];
  const float* Wk    = (const float*)d_in[7];
  const float* bk    = (const float*)d_in[8];
  const float* gk    = (const float*)d_in[9];
  const float* betak = (const float*)d_in[10];
  float* out = (float*)d_out;

  float* ws    = (float*)d_ws;
  float* qp    = ws;                       // 32*512      = 16384 floats
  float* kp    = ws + 16384;               // 1024*512    = 524288 floats
  float* masks = ws + 16384 + 524288;      // 32*8*32     = 8192 floats

  // q projection: [32,512] = Q @ Wq^T + bq
  gemm_wmma_f32<<<dim3(2, 32), 32, 0, stream>>>(Q, Wq, bq, qp);
  // k projection: [1024,512] = K @ Wk^T + bk
  gemm_wmma_f32<<<dim3(64, 32), 32, 0, stream>>>(K, Wk, bk, kp);
  // layernorms
  ln_kernel<<<32, 256, 0, stream>>>(qp, gq, betaq);
  ln_kernel<<<1024, 256, 0, stream>>>(kp, gk, betak);
  // logits + softmax + gumbel-softmax masks
  mask_kernel<<<32, 32, 0, stream>>>(qp, kp, masks);
  // soft gather (dominant, bandwidth-bound): 32768 float4 per batch
  select_kernel<<<dim3(128, 32), 256, 0, stream>>>(masks, V, out);
}